// GINE_2095944040567
// MI455X (gfx1250) — compile-verified
//
#include <hip/hip_runtime.h>
#include <hip/hip_bf16.h>

// ---------------------------------------------------------------------------
// GINEConv + MLP(BN,ReLU) for MI455X (gfx1250), wave32, V_WMMA_F32_16X16X4_F32
// Sizes: N=50000 nodes, E=1e6 edges, d=64, edge_dim=16.  N%16==0, E%16==0.
// ---------------------------------------------------------------------------

typedef float v2f __attribute__((ext_vector_type(2)));
typedef float v8f __attribute__((ext_vector_type(8)));

#define D 64
#define ED 16
#define BN_EPS 1e-5f

// D = A(16x4 f32) * B(4x16 f32) + C(16x16 f32), chained accumulate.
static __device__ __forceinline__ v8f wmma4(v2f a, v2f b, v8f c) {
  return __builtin_amdgcn_wmma_f32_16x16x4_f32(false, a, false, b, (short)0, c,
                                               false, false);
}

// ---------------------------------------------------------------------------
// K0: agg <- x  (GIN eps = 0 means h0 = x + sum(msg); seed accumulator with x)
//     stats/scale-shift area <- 0
// ---------------------------------------------------------------------------
__global__ void k_init(const float* __restrict__ x, float* __restrict__ agg,
                       float* __restrict__ stats, int total) {
  int i = blockIdx.x * blockDim.x + threadIdx.x;
  if (i < total) agg[i] = x[i];
  if (i < 256) stats[i] = 0.0f;
}

// ---------------------------------------------------------------------------
// K1: per 16-edge tile (one wave):
//   ea[16x64] = edge_attr[16x16] @ W_edge^T   via 4 Kslices x 4 Ntiles WMMA
//   msg = relu(x[src] + ea + b_edge);  atomicAdd into agg[dst]
// A layout (16x4 f32): lane l -> row M = l&15, VGPR r -> K = 2*(l>>4)+r
// B layout (4x16 f32): lane l -> col N = l&15, VGPR r -> K = 2*(l>>4)+r
// C layout (16x16 f32): lane l -> col N = l&15, VGPR r -> M = r + 8*(l>>4)
//
// The 32 x[src] gathers are issued BEFORE the WMMA chain so their L2 latency
// overlaps the matrix work instead of serializing after it.
// ---------------------------------------------------------------------------
__global__ void __launch_bounds__(256)
k_edge(const float* __restrict__ x, const long long* __restrict__ ei,
       const float* __restrict__ ea, const float* __restrict__ We,
       const float* __restrict__ be, float* __restrict__ agg, int nE) {
  __shared__ float sW[D * ED];  // W_edge[n][k], row-major [64][16]
  __shared__ float sb[D];
  for (int i = threadIdx.x; i < D * ED; i += blockDim.x) sW[i] = We[i];
  if (threadIdx.x < D) sb[threadIdx.x] = be[threadIdx.x];
  __syncthreads();

  const int wave = threadIdx.x >> 5;
  const int lane = threadIdx.x & 31;
  const int tile = blockIdx.x * (blockDim.x >> 5) + wave;
  const int ntiles = nE >> 4;
  if (tile >= ntiles) return;

  const int e0    = tile << 4;
  const int mrow  = lane & 15;        // A row / B,C column within tile
  const int khalf = (lane >> 4) << 1; // 0 or 2: K sub-pair owned by this lane
  const int moff  = (lane >> 4) << 3; // 0 or 8: C row offset for this lane

  // Prefetch the next block-worth of edge_attr rows (global_prefetch_b8).
  __builtin_prefetch(ea + (size_t)(e0 + 128 + mrow) * ED, 0, 0);

  // ---- 1) edge indices -> row base pointers (compute i64 math once) -------
  const float* xs[8];  // x row of message source, per C-row this lane holds
  float*       ad[8];  // agg row of aggregation target
#pragma unroll
  for (int r = 0; r < 8; ++r) {
    const int e = e0 + r + moff;
    xs[r] = x + (size_t)((int)ei[e]) * D;
    ad[r] = agg + (size_t)((int)ei[nE + e]) * D;
  }

  // ---- 2) issue all 32 gathers up front ------------------------------------
  float xv[4][8];
#pragma unroll
  for (int nb = 0; nb < 4; ++nb) {
    const int n = nb * 16 + mrow;
#pragma unroll
    for (int r = 0; r < 8; ++r) xv[nb][r] = xs[r][n];
  }

  // ---- 3) WMMA chain: ea tile = edge_attr @ W_edge^T -----------------------
  v8f acc[4] = {};  // 4 N-tiles of 16 features each
  const float* arow = ea + (size_t)(e0 + mrow) * ED;
#pragma unroll
  for (int j = 0; j < 4; ++j) {  // K = 16 -> 4 slices of 4
    const int k0 = 4 * j + khalf;
    v2f a = *(const v2f*)(arow + k0);
#pragma unroll
    for (int nb = 0; nb < 4; ++nb) {
      v2f b = *(const v2f*)&sW[(nb * 16 + mrow) * ED + k0];
      acc[nb] = wmma4(a, b, acc[nb]);
    }
  }

  // ---- 4) msg = relu(x_j + ea + b_edge), then one batched atomic burst -----
  float msg[4][8];
#pragma unroll
  for (int nb = 0; nb < 4; ++nb) {
    const float bias = sb[nb * 16 + mrow];
#pragma unroll
    for (int r = 0; r < 8; ++r) {
      float v = xv[nb][r] + acc[nb][r] + bias;
      msg[nb][r] = v > 0.0f ? v : 0.0f;
    }
  }
#pragma unroll
  for (int nb = 0; nb < 4; ++nb) {
    const int n = nb * 16 + mrow;
#pragma unroll
    for (int r = 0; r < 8; ++r) atomicAdd(ad[r] + n, msg[nb][r]);
  }
}

// ---------------------------------------------------------------------------
// K2: h1 = h0 @ W1^T + b1 ; accumulate per-column sum & sumsq for BatchNorm
// ---------------------------------------------------------------------------
__global__ void __launch_bounds__(256)
k_gemm1(const float* __restrict__ h0, const float* __restrict__ W1,
        const float* __restrict__ b1, float* __restrict__ h1,
        float* __restrict__ stats, int nRows) {
  __shared__ float sW[D * D];  // W1[n][k] row-major [64][64]
  __shared__ float sb[D];
  __shared__ float sred[2 * D];
  for (int i = threadIdx.x; i < D * D; i += blockDim.x) sW[i] = W1[i];
  if (threadIdx.x < D) sb[threadIdx.x] = b1[threadIdx.x];
  if (threadIdx.x < 2 * D) sred[threadIdx.x] = 0.0f;
  __syncthreads();

  const int wave = threadIdx.x >> 5;
  const int lane = threadIdx.x & 31;
  const int tile = blockIdx.x * (blockDim.x >> 5) + wave;
  const int ntiles = nRows >> 4;
  const int mrow = lane & 15, khalf = (lane >> 4) << 1, moff = (lane >> 4) << 3;

  if (tile < ntiles) {
    const int m0 = tile << 4;
    v8f acc[4] = {};
    const float* arow = h0 + (size_t)(m0 + mrow) * D;
#pragma unroll
    for (int j = 0; j < 16; ++j) {  // K = 64 -> 16 slices of 4
      const int k0 = 4 * j + khalf;
      v2f a = *(const v2f*)(arow + k0);
#pragma unroll
      for (int nb = 0; nb < 4; ++nb) {
        v2f b = *(const v2f*)&sW[(nb * 16 + mrow) * D + k0];
        acc[nb] = wmma4(a, b, acc[nb]);
      }
    }
#pragma unroll
    for (int nb = 0; nb < 4; ++nb) {
      const int n = nb * 16 + mrow;
      const float bias = sb[n];
      float s = 0.0f, sq = 0.0f;
#pragma unroll
      for (int r = 0; r < 8; ++r) {
        const int m = m0 + r + moff;
        const float v = acc[nb][r] + bias;
        h1[(size_t)m * D + n] = v;
        s += v;
        sq += v * v;
      }
      atomicAdd(&sred[n], s);  // ds_add_f32 block-level reduction
      atomicAdd(&sred[D + n], sq);
    }
  }
  __syncthreads();
  if (threadIdx.x < 2 * D) atomicAdd(&stats[threadIdx.x], sred[threadIdx.x]);
}

// ---------------------------------------------------------------------------
// K3: fold BN into per-column scale/shift:  y = sc*h + sh
// ---------------------------------------------------------------------------
__global__ void k_stats(const float* __restrict__ gamma,
                        const float* __restrict__ beta,
                        const float* __restrict__ stats,
                        float* __restrict__ ss, float invN) {
  const int n = threadIdx.x;
  if (n < D) {
    const float mean = stats[n] * invN;
    const float var = stats[D + n] * invN - mean * mean;  // biased var
    const float inv = rsqrtf(var + BN_EPS);
    const float sc = gamma[n] * inv;
    ss[n] = sc;
    ss[D + n] = beta[n] - sc * mean;
  }
}

// ---------------------------------------------------------------------------
// K4: out = relu(BN(h1)) @ W2^T + b2 ; BN+ReLU fused into the A-load
// ---------------------------------------------------------------------------
__global__ void __launch_bounds__(256)
k_gemm2(const float* __restrict__ h1, const float* __restrict__ W2,
        const float* __restrict__ b2, const float* __restrict__ ss,
        float* __restrict__ out, int nRows) {
  __shared__ float sW[D * D];
  __shared__ float sb[D];
  __shared__ float ssc[D], ssh[D];
  for (int i = threadIdx.x; i < D * D; i += blockDim.x) sW[i] = W2[i];
  if (threadIdx.x < D) {
    sb[threadIdx.x] = b2[threadIdx.x];
    ssc[threadIdx.x] = ss[threadIdx.x];
    ssh[threadIdx.x] = ss[D + threadIdx.x];
  }
  __syncthreads();

  const int wave = threadIdx.x >> 5;
  const int lane = threadIdx.x & 31;
  const int tile = blockIdx.x * (blockDim.x >> 5) + wave;
  const int ntiles = nRows >> 4;
  if (tile >= ntiles) return;

  const int m0 = tile << 4;
  const int mrow = lane & 15, khalf = (lane >> 4) << 1, moff = (lane >> 4) << 3;

  v8f acc[4] = {};
  const float* arow = h1 + (size_t)(m0 + mrow) * D;
#pragma unroll
  for (int j = 0; j < 16; ++j) {
    const int k0 = 4 * j + khalf;
    v2f h = *(const v2f*)(arow + k0);
    v2f a;
    a.x = fmaxf(ssc[k0] * h.x + ssh[k0], 0.0f);
    a.y = fmaxf(ssc[k0 + 1] * h.y + ssh[k0 + 1], 0.0f);
#pragma unroll
    for (int nb = 0; nb < 4; ++nb) {
      v2f b = *(const v2f*)&sW[(nb * 16 + mrow) * D + k0];
      acc[nb] = wmma4(a, b, acc[nb]);
    }
  }
#pragma unroll
  for (int nb = 0; nb < 4; ++nb) {
    const int n = nb * 16 + mrow;
    const float bias = sb[n];
#pragma unroll
    for (int r = 0; r < 8; ++r) {
      const int m = m0 + r + moff;
      out[(size_t)m * D + n] = acc[nb][r] + bias;
    }
  }
}

// ---------------------------------------------------------------------------
extern "C" void kernel_launch(void* const* d_in, const int* in_sizes, int n_in,
                              void* d_out, int out_size, void* d_ws,
                              size_t ws_size, hipStream_t stream) {
  const float*     x      = (const float*)d_in[0];
  const long long* ei     = (const long long*)d_in[1];  // int64 [2, E]
  const float*     eattr  = (const float*)d_in[2];
  const float*     W_edge = (const float*)d_in[3];
  const float*     b_edge = (const float*)d_in[4];
  const float*     W1     = (const float*)d_in[5];
  const float*     b1     = (const float*)d_in[6];
  const float*     gamma  = (const float*)d_in[7];
  const float*     beta   = (const float*)d_in[8];
  const float*     W2     = (const float*)d_in[9];
  const float*     b2     = (const float*)d_in[10];
  float*           out    = (float*)d_out;

  const int N = in_sizes[0] / D;   // 50000
  const int E = in_sizes[2] / ED;  // 1000000

  // Workspace layout (floats): agg/h0 [N*64] | h1 [N*64] | stats[128] | ss[128]
  float* ws    = (float*)d_ws;
  float* agg   = ws;
  float* h1    = ws + (size_t)N * D;
  float* stats = ws + (size_t)2 * N * D;
  float* ss    = stats + 2 * D;

  const int total = N * D;
  k_init<<<(total + 255) / 256, 256, 0, stream>>>(x, agg, stats, total);

  const int etiles = E >> 4;  // wave-tiles of 16 edges
  k_edge<<<(etiles + 7) / 8, 256, 0, stream>>>(x, ei, eattr, W_edge, b_edge,
                                               agg, E);

  const int ntiles = N >> 4;
  k_gemm1<<<(ntiles + 7) / 8, 256, 0, stream>>>(agg, W1, b1, h1, stats, N);

  k_stats<<<1, 64, 0, stream>>>(gamma, beta, stats, ss, 1.0f / (float)N);

  k_gemm2<<<(ntiles + 7) / 8, 256, 0, stream>>>(h1, W2, b2, ss, out, N);
}